// ChaoticEnricher_43224550867184
// MI455X (gfx1250) — compile-verified
//
#include <hip/hip_runtime.h>
#include <math.h>

#define LDT 0.01f
#define LSTEPS 30
#define DIM 2048
#define FSTRIDE 20   // 18 features padded to 20 = 5 WMMA K4 chunks

typedef float v2f __attribute__((ext_vector_type(2)));
typedef float v8f __attribute__((ext_vector_type(8)));

// Branch-free tanh: t = exp(-2|x|) (v_exp_f32), r = (1-t)/(1+t), restore sign.
// Exact saturation for large |x|; ~few-ulp absolute error elsewhere; no EXEC
// divergence (ocml tanhf emits a two-path s_and_saveexec expansion).
__device__ __forceinline__ float tanh_fast(float v) {
  float t = __expf(-2.0f * fabsf(v));
  float r = (1.0f - t) * __builtin_amdgcn_rcpf(1.0f + t);
  return __builtin_copysignf(r, v);
}

// ---------------- Kernel 0: zero-pad W_out [D,18] -> Wp [D,20] ----------------
__global__ void k_padW(const float* __restrict__ W_out, float* __restrict__ Wp) {
  int gid = blockIdx.x * blockDim.x + threadIdx.x;   // D*20 elements
  if (gid >= DIM * FSTRIDE) return;
  int row = gid / FSTRIDE;
  int col = gid % FSTRIDE;
  Wp[gid] = (col < 18) ? W_out[row * 18 + col] : 0.0f;
}

// ---------------- Kernel 1: state0 = x @ W_in^T + b_in  (one wave per row) ----------------
__global__ void k_state0(const float* __restrict__ x,
                         const float* __restrict__ W_in,
                         const float* __restrict__ b_in,
                         float* __restrict__ state0, int N) {
  int gid  = blockIdx.x * blockDim.x + threadIdx.x;
  int row  = gid >> 5;
  int lane = gid & 31;
  if (row >= N) return;
  const float4* xr = (const float4*)(x + (size_t)row * DIM);
  const float4* w0 = (const float4*)(W_in);
  const float4* w1 = (const float4*)(W_in + DIM);
  const float4* w2 = (const float4*)(W_in + 2 * DIM);
  float s0 = 0.f, s1 = 0.f, s2 = 0.f;
#pragma unroll 4
  for (int i = lane; i < DIM / 4; i += 32) {
    float4 xv = xr[i];
    float4 a = w0[i], b = w1[i], c = w2[i];
    s0 = fmaf(xv.x, a.x, fmaf(xv.y, a.y, fmaf(xv.z, a.z, fmaf(xv.w, a.w, s0))));
    s1 = fmaf(xv.x, b.x, fmaf(xv.y, b.y, fmaf(xv.z, b.z, fmaf(xv.w, b.w, s1))));
    s2 = fmaf(xv.x, c.x, fmaf(xv.y, c.y, fmaf(xv.z, c.z, fmaf(xv.w, c.w, s2))));
  }
#pragma unroll
  for (int off = 16; off > 0; off >>= 1) {
    s0 += __shfl_xor(s0, off, 32);
    s1 += __shfl_xor(s1, off, 32);
    s2 += __shfl_xor(s2, off, 32);
  }
  if (lane == 0) {
    state0[(size_t)row * 4 + 0] = s0 + b_in[0];
    state0[(size_t)row * 4 + 1] = s1 + b_in[1];
    state0[(size_t)row * 4 + 2] = s2 + b_in[2];
  }
}

// ---------------- Kernel 2: Lorenz RK4 + trajectory stats (one thread per row) ----------------
__device__ __forceinline__ void lderiv(float sx, float sy, float sz,
                                       float sg, float rh, float bt,
                                       float& dx, float& dy, float& dz) {
  dx = sg * (sy - sx);
  dy = sx * (rh - sz) - sy;
  dz = sx * sy - bt * sz;
}

__device__ __forceinline__ void rk4(float& sx, float& sy, float& sz,
                                    float sg, float rh, float bt) {
  float k1x, k1y, k1z, k2x, k2y, k2z, k3x, k3y, k3z, k4x, k4y, k4z;
  lderiv(sx, sy, sz, sg, rh, bt, k1x, k1y, k1z);
  lderiv(sx + 0.5f * LDT * k1x, sy + 0.5f * LDT * k1y, sz + 0.5f * LDT * k1z,
         sg, rh, bt, k2x, k2y, k2z);
  lderiv(sx + 0.5f * LDT * k2x, sy + 0.5f * LDT * k2y, sz + 0.5f * LDT * k2z,
         sg, rh, bt, k3x, k3y, k3z);
  lderiv(sx + LDT * k3x, sy + LDT * k3y, sz + LDT * k3z,
         sg, rh, bt, k4x, k4y, k4z);
  const float c = LDT / 6.0f;
  sx += c * (k1x + 2.f * k2x + 2.f * k3x + k4x);
  sy += c * (k1y + 2.f * k2y + 2.f * k3y + k4y);
  sz += c * (k1z + 2.f * k2z + 2.f * k3z + k4z);
}

__global__ void k_ode(const float* __restrict__ state0,
                      const float* __restrict__ lorenz,
                      float* __restrict__ feats, int N) {
  int row = blockIdx.x * blockDim.x + threadIdx.x;
  if (row >= N) return;
  const float sg = fmaxf(fabsf(lorenz[0]), 0.1f);
  const float rh = fmaxf(fabsf(lorenz[1]), 0.1f);
  const float bt = fmaxf(fabsf(lorenz[2]), 0.1f);
  const float ix = state0[(size_t)row * 4 + 0];
  const float iy = state0[(size_t)row * 4 + 1];
  const float iz = state0[(size_t)row * 4 + 2];

  // pass 1: sum / min / max / final
  float sx = ix, sy = iy, sz = iz;
  float sumx = ix, sumy = iy, sumz = iz;
  float minx = ix, miny = iy, minz = iz;
  float maxx = ix, maxy = iy, maxz = iz;
  for (int t = 0; t < LSTEPS; ++t) {
    rk4(sx, sy, sz, sg, rh, bt);
    sumx += sx; sumy += sy; sumz += sz;
    minx = fminf(minx, sx); miny = fminf(miny, sy); minz = fminf(minz, sz);
    maxx = fmaxf(maxx, sx); maxy = fmaxf(maxy, sy); maxz = fmaxf(maxz, sz);
  }
  const float fx = sx, fy = sy, fz = sz;
  const float inv = 1.0f / (float)(LSTEPS + 1);
  const float mex = sumx * inv, mey = sumy * inv, mez = sumz * inv;

  // pass 2: exact two-pass variance by deterministic recompute (ddof=1)
  sx = ix; sy = iy; sz = iz;
  float vx = (sx - mex) * (sx - mex);
  float vy = (sy - mey) * (sy - mey);
  float vz = (sz - mez) * (sz - mez);
  for (int t = 0; t < LSTEPS; ++t) {
    rk4(sx, sy, sz, sg, rh, bt);
    vx += (sx - mex) * (sx - mex);
    vy += (sy - mey) * (sy - mey);
    vz += (sz - mez) * (sz - mez);
  }
  const float dn = 1.0f / (float)LSTEPS;  // n-1 = 30
  float* f = feats + (size_t)row * FSTRIDE;
  f[0]  = ix;  f[1]  = iy;  f[2]  = iz;
  f[3]  = fx;  f[4]  = fy;  f[5]  = fz;
  f[6]  = mex; f[7]  = mey; f[8]  = mez;
  f[9]  = sqrtf(vx * dn); f[10] = sqrtf(vy * dn); f[11] = sqrtf(vz * dn);
  f[12] = minx; f[13] = miny; f[14] = minz;
  f[15] = maxx; f[16] = maxy; f[17] = maxz;
  f[18] = 0.f;  f[19] = 0.f;
}

// ---------------- Kernel 3: out = x + tanh(feats @ W_out^T + b_out)*|s| ----------------
// WMMA F32_16X16X4_F32. One wave computes a 16x64 output tile (4 column
// sub-tiles reusing one set of 5 A fragments; 20 WMMAs per wave).
// A layout (16x4 f32): lane&15 = M, (lane>>4)*2 = K offset, VGPR0/1 = K,K+1.
// B layout (4x16 f32): lane&15 = N column, (lane>>4)*2 = K offset (mirror of A).
// C/D layout: lane&15 = N, acc[v] -> M = v + 8*(lane>>4).
// A/B fragments are loaded into distinct live registers BEFORE the WMMA chain
// so the loads form one s_clause and overlap the matrix ops (avoids the
// load->wait0->wmma serialization the allocator produced when recycling one
// register pair).
__global__ void k_gemm2(const float* __restrict__ feats,   // [N, 20] padded
                        const float* __restrict__ Wp,      // [D, 20] zero-padded
                        const float* __restrict__ b_out,   // [D]
                        const float* __restrict__ x,
                        const float* __restrict__ strength_p,
                        float* __restrict__ out, int N) {
  int gid  = blockIdx.x * blockDim.x + threadIdx.x;
  int wid  = gid >> 5;
  int lane = gid & 31;
  const int ctiles = DIM / 64;                 // 32 column super-tiles
  const int tiles  = (N / 16) * ctiles;
  if (wid >= tiles) return;                    // wave-uniform: EXEC stays all-1s
  const int r0 = (wid / ctiles) * 16;
  const int c0 = (wid % ctiles) * 64;
  const int m  = lane & 15;
  const int kh = (lane >> 4) * 2;

  // Load the 5 A fragments once (reused by all 4 column sub-tiles).
  // feats + (r0+m)*20 + kh is 8-byte aligned (both terms even) -> b64 loads.
  v2f a[5];
  const v2f* fr = (const v2f*)(feats + (size_t)(r0 + m) * FSTRIDE + kh);
#pragma unroll
  for (int kc = 0; kc < 5; ++kc) a[kc] = fr[kc * 2];

  const float st = fabsf(*strength_p);
#pragma unroll
  for (int ct = 0; ct < 4; ++ct) {
    const int c   = c0 + ct * 16;
    const int col = c + m;

    // All 5 B fragments in distinct registers -> one clause, pipelined waits.
    v2f b[5];
    const v2f* wr = (const v2f*)(Wp + (size_t)col * FSTRIDE + kh);
#pragma unroll
    for (int kc = 0; kc < 5; ++kc) b[kc] = wr[kc * 2];

    v8f acc = {};
#pragma unroll
    for (int kc = 0; kc < 5; ++kc) {
      // (neg_a, A, neg_b, B, c_mod, C, reuse_a, reuse_b)
      acc = __builtin_amdgcn_wmma_f32_16x16x4_f32(false, a[kc], false, b[kc],
                                                  (short)0, acc, false, false);
    }
    const float bo = b_out[col];
#pragma unroll
    for (int v = 0; v < 8; ++v) {
      const int row = r0 + v + 8 * (lane >> 4);
      const size_t idx = (size_t)row * DIM + col;
      out[idx] = x[idx] + tanh_fast(acc[v] + bo) * st;
    }
  }
}

extern "C" void kernel_launch(void* const* d_in, const int* in_sizes, int n_in,
                              void* d_out, int out_size, void* d_ws, size_t ws_size,
                              hipStream_t stream) {
  const float* x        = (const float*)d_in[0];
  const float* lorenz   = (const float*)d_in[1];
  const float* strength = (const float*)d_in[2];
  const float* W_in     = (const float*)d_in[3];
  const float* b_in     = (const float*)d_in[4];
  const float* W_out    = (const float*)d_in[5];
  const float* b_out    = (const float*)d_in[6];
  float* out = (float*)d_out;

  const int N = in_sizes[0] / DIM;            // 16384 rows

  float* state0 = (float*)d_ws;                            // N*4 floats
  float* feats  = (float*)d_ws + (size_t)N * 4;            // N*20 floats
  float* Wp     = (float*)d_ws + (size_t)N * 4 + (size_t)N * FSTRIDE;  // D*20 floats

  // K0: zero-pad W_out -> Wp [D,20]
  {
    int n = DIM * FSTRIDE;
    k_padW<<<(n + 255) / 256, 256, 0, stream>>>(W_out, Wp);
  }
  // K1: one wave per row -> N*32 threads
  {
    int threads = N * 32;
    k_state0<<<(threads + 255) / 256, 256, 0, stream>>>(x, W_in, b_in, state0, N);
  }
  // K2: one thread per row
  {
    k_ode<<<(N + 255) / 256, 256, 0, stream>>>(state0, lorenz, feats, N);
  }
  // K3: one wave per 16x64 tile -> (N/16)*(DIM/64) waves
  {
    int tiles = (N / 16) * (DIM / 64);
    int threads = tiles * 32;
    k_gemm2<<<(threads + 255) / 256, 256, 0, stream>>>(feats, Wp, b_out, x,
                                                       strength, out, N);
  }
}